// LightGCNPredictor_5574867550503
// MI455X (gfx1250) — compile-verified
//
#include <hip/hip_runtime.h>

#define NUM_USERS 100000
#define NUM_ITEMS 50000
#define N_NODES   (NUM_USERS + NUM_ITEMS)
#define EMB_DIM   64
#define NUM_LAYERS 3
#define ALPHA     (1.0f / (NUM_LAYERS + 1))

typedef __attribute__((ext_vector_type(2))) float v2f;
typedef __attribute__((ext_vector_type(8))) float v8f;

// ---------------- utility kernels ----------------

__global__ void zero_f32_kernel(float* __restrict__ p, long n) {
  long i = (long)blockIdx.x * blockDim.x + threadIdx.x;
  long stride = (long)gridDim.x * blockDim.x;
  for (; i < n; i += stride) p[i] = 0.0f;
}

__global__ void deg_kernel(const int* __restrict__ col, float* __restrict__ deg, int E) {
  int e = blockIdx.x * blockDim.x + threadIdx.x;
  if (e < E) atomicAdd(&deg[col[e]], 1.0f);
}

__global__ void dinv_kernel(const float* __restrict__ deg, float* __restrict__ dinv, int n) {
  int i = blockIdx.x * blockDim.x + threadIdx.x;
  if (i < n) {
    float d = deg[i];
    dinv[i] = (d > 0.0f) ? rsqrtf(d) : 0.0f;
  }
}

// out_acc = alpha * emb ; x_cur = emb
__global__ void init_kernel(const float* __restrict__ emb, float* __restrict__ out_acc,
                            float* __restrict__ x_cur, long n) {
  long i = (long)blockIdx.x * blockDim.x + threadIdx.x;
  long stride = (long)gridDim.x * blockDim.x;
  for (; i < n; i += stride) {
    float v = emb[i];
    out_acc[i] = v * ALPHA;
    x_cur[i]   = v;
  }
}

// One thread per (edge, 4-float chunk): 16 chunk-threads cover EMB_DIM=64.
// Gathers x[row] (L2-resident), scales by dinv[row]*dinv[col], scatters with
// f32 atomics into x_new[col] (also L2-resident: 38.4MB target << 192MB L2).
__global__ void scatter_kernel(const int* __restrict__ row, const int* __restrict__ col,
                               const float* __restrict__ dinv,
                               const float* __restrict__ x, float* __restrict__ xn, int E) {
  long tid = (long)blockIdx.x * blockDim.x + threadIdx.x;
  if (tid >= (long)E * 16) return;
  int e = (int)(tid >> 4);
  int c = (int)(tid & 15);
  int r  = row[e];
  int cn = col[e];
  float w = dinv[r] * dinv[cn];
  const float4* xv = (const float4*)(x + (long)r * EMB_DIM);
  float4 v = xv[c];
  float* dst = xn + (long)cn * EMB_DIM + c * 4;
  atomicAdd(dst + 0, w * v.x);
  atomicAdd(dst + 1, w * v.y);
  atomicAdd(dst + 2, w * v.z);
  atomicAdd(dst + 3, w * v.w);
}

// out_acc += alpha * x_new
__global__ void axpy_kernel(const float* __restrict__ xn, float* __restrict__ out_acc, long n) {
  long i = (long)blockIdx.x * blockDim.x + threadIdx.x;
  long stride = (long)gridDim.x * blockDim.x;
  for (; i < n; i += stride) out_acc[i] += ALPHA * xn[i];
}

// ---------------- scores = user_emb[src] @ item_emb^T via V_WMMA_F32_16X16X4_F32 ----
//
// One wave computes one 16x16 output tile, stepping K=64 in chunks of 4.
// A (16x4 f32): lane L<16 holds (M=L, K=kk+{0,1}); lane L>=16 holds (M=L-16, K=kk+{2,3}).
// B (4x16 f32): symmetric — lane L<16 holds (N=L, K=kk+{0,1}); upper half K=kk+{2,3}.
// Since B[k,n] = item_emb[n][k], both fragments are plain row-major b64 loads.
// C/D (16x16 f32, v8f): VGPR v -> M=v (lanes 0-15) / M=v+8 (lanes 16-31), N=lane&15.
__global__ void scores_wmma_kernel(const float* __restrict__ nodes,
                                   const int* __restrict__ src,
                                   float* __restrict__ scores) {
  const int lane = threadIdx.x & 31;
  const int wave = threadIdx.x >> 5;
  const int tileM = blockIdx.y;                       // 0..63   (1024/16)
  const int tileN = blockIdx.x * 8 + wave;            // 0..3124 (50000/16)
  if (tileN >= NUM_ITEMS / 16) return;                // wave-uniform, EXEC stays all-1s

  const int half = lane >> 4;                         // K sub-pair selector
  const int l    = lane & 15;

  const float* U = nodes;                             // user rows = node rows [0, NUM_USERS)
  const float* I = nodes + (long)NUM_USERS * EMB_DIM; // item rows

  const int   urow = src[tileM * 16 + l];
  const float* arow = U + (long)urow * EMB_DIM + half * 2;
  const int   icol = tileN * 16 + l;
  const float* brow = I + (long)icol * EMB_DIM + half * 2;

  v8f c = {};
#pragma unroll
  for (int kk = 0; kk < EMB_DIM / 4; ++kk) {
    v2f a = *(const v2f*)(arow + kk * 4);             // 8B-aligned b64 load
    v2f b = *(const v2f*)(brow + kk * 4);
    c = __builtin_amdgcn_wmma_f32_16x16x4_f32(
        /*neg_a=*/false, a, /*neg_b=*/false, b,
        /*c_mod=*/(short)0, c, /*reuse_a=*/false, /*reuse_b=*/false);
  }

  const long base = (long)(tileM * 16) * NUM_ITEMS + (long)tileN * 16 + l;
#pragma unroll
  for (int v = 0; v < 8; ++v) {
    int M = v + half * 8;
    scores[base + (long)M * NUM_ITEMS] = c[v];
  }
}

// ---------------- host launch ----------------

extern "C" void kernel_launch(void* const* d_in, const int* in_sizes, int n_in,
                              void* d_out, int out_size, void* d_ws, size_t ws_size,
                              hipStream_t stream) {
  const int*   edge_index = (const int*)d_in[0];   // (2, E) int32
  const int*   src        = (const int*)d_in[1];   // (1024,) int32
  const float* emb        = (const float*)d_in[2]; // (N_NODES, 64) f32
  float*       scores     = (float*)d_out;         // (1024, 50000) f32

  const int E = in_sizes[0] / 2;
  const int* row = edge_index;
  const int* col = edge_index + E;

  // workspace layout (floats): deg | dinv | bufA | bufB | out_acc  (~117 MB)
  float* deg     = (float*)d_ws;
  float* dinv    = deg  + N_NODES;
  float* bufA    = dinv + N_NODES;
  float* bufB    = bufA + (long)N_NODES * EMB_DIM;
  float* out_acc = bufB + (long)N_NODES * EMB_DIM;
  const long NE  = (long)N_NODES * EMB_DIM;

  zero_f32_kernel<<<1024, 256, 0, stream>>>(deg, (long)N_NODES);
  deg_kernel<<<(E + 255) / 256, 256, 0, stream>>>(col, deg, E);
  dinv_kernel<<<(N_NODES + 255) / 256, 256, 0, stream>>>(deg, dinv, N_NODES);
  init_kernel<<<4096, 256, 0, stream>>>(emb, out_acc, bufA, NE);

  float* x_cur = bufA;
  float* x_new = bufB;
  for (int layer = 0; layer < NUM_LAYERS; ++layer) {
    zero_f32_kernel<<<4096, 256, 0, stream>>>(x_new, NE);
    long threads = (long)E * 16;
    scatter_kernel<<<(int)((threads + 255) / 256), 256, 0, stream>>>(row, col, dinv, x_cur, x_new, E);
    axpy_kernel<<<4096, 256, 0, stream>>>(x_new, out_acc, NE);
    float* t = x_cur; x_cur = x_new; x_new = t;
  }

  dim3 grid((NUM_ITEMS / 16 + 7) / 8, 1024 / 16);   // 391 x 64 blocks, 8 waves/block
  scores_wmma_kernel<<<grid, 256, 0, stream>>>(out_acc, src, scores);
}